// ClustGeoEdgeEncoderVectorized_61555471286690
// MI455X (gfx1250) — compile-verified
//
#include <hip/hip_runtime.h>

typedef __attribute__((ext_vector_type(2))) float v2f;
typedef __attribute__((ext_vector_type(8))) float v8f;

#define S_PER_CLUST 32  // reference: N_VOX=8192, N_CLUST=256 -> S=32

__device__ __forceinline__ unsigned int f32_bits(float x) {
    union { float f; unsigned int u; } c; c.f = x; return c.u;
}
__device__ __forceinline__ float bits_f32(unsigned int u) {
    union { float f; unsigned int u; } c; c.u = u; return c.f;
}

__global__ __launch_bounds__(256)
void clust_geo_edge_wmma(const float* __restrict__ data,
                         const int* __restrict__ edge_index,
                         float* __restrict__ out,
                         int n_edges)
{
    const int lane = threadIdx.x & 31;
    const int wave = threadIdx.x >> 5;
    const int e    = blockIdx.x * 8 + wave;
    if (e >= n_edges) return;  // wave-uniform guard

    const int ci = edge_index[e];
    const int cj = edge_index[n_edges + e];

    const float4* dv = (const float4*)data;

    // ---- diagonal pair: reference yields exact ties at dist 0 for every self
    // pair, argmin picks flat index 0 -> v1 = v2 = voxels[ci*S], all else 0.
    if (ci == cj) {
        const float4 p = dv[ci * S_PER_CLUST];
        float f = 0.0f;
        switch (lane) {
            case 0: case 3: f = p.x; break;
            case 1: case 4: f = p.y; break;
            case 2: case 5: f = p.z; break;
            default: break;   // disp = 0, lend = 0, B = 0
        }
        if (lane < 19) out[(size_t)e * 19 + lane] = f;
        return;
    }

    const int  n  = lane & 15;          // column / row-within-tile index
    const bool hi = (lane & 16) != 0;   // upper half-wave carries K=2,3

    // data rows are [x, y, z, batch] -> 16B aligned float4
    const float4 ra0 = dv[ci * S_PER_CLUST + n];
    const float4 ra1 = dv[ci * S_PER_CLUST + 16 + n];
    const float4 rb0 = dv[cj * S_PER_CLUST + n];
    const float4 rb1 = dv[cj * S_PER_CLUST + 16 + n];

    const float sqa0 = ra0.x*ra0.x + ra0.y*ra0.y + ra0.z*ra0.z;
    const float sqa1 = ra1.x*ra1.x + ra1.y*ra1.y + ra1.z*ra1.z;
    const float sqb0 = rb0.x*rb0.x + rb0.y*rb0.y + rb0.z*rb0.z;
    const float sqb1 = rb1.x*rb1.x + rb1.y*rb1.y + rb1.z*rb1.z;

    // A 16x4 (M=rows of cluster i): lanes 0-15 hold K=0,1 = (x,y); lanes 16-31 K=2,3 = (z, |a|^2)
    v2f A0, A1, B0, B1;
    A0[0] = hi ? ra0.z : ra0.x;   A0[1] = hi ? sqa0 : ra0.y;
    A1[0] = hi ? ra1.z : ra1.x;   A1[1] = hi ? sqa1 : ra1.y;
    // B 4x16 (N=rows of cluster j as columns): K=0..2 = -2*(x,y,z), K=3 = 1
    B0[0] = hi ? (-2.0f*rb0.z) : (-2.0f*rb0.x);   B0[1] = hi ? 1.0f : (-2.0f*rb0.y);
    B1[0] = hi ? (-2.0f*rb1.z) : (-2.0f*rb1.x);   B1[1] = hi ? 1.0f : (-2.0f*rb1.y);

    // C accumulator = |b|^2 broadcast along M (lane-local: lane fixes N)
    const v8f c0 = { sqb0, sqb0, sqb0, sqb0, sqb0, sqb0, sqb0, sqb0 };
    const v8f c1 = { sqb1, sqb1, sqb1, sqb1, sqb1, sqb1, sqb1, sqb1 };

    // D = A*B + C = |a|^2 + |b|^2 - 2 a.b  == squared distance, per 16x16 tile
    // args: (neg_a, A, neg_b, B, c_mod, C, reuse_a, reuse_b)
    v8f D00 = __builtin_amdgcn_wmma_f32_16x16x4_f32(false, A0, false, B0, (short)0, c0, false, false);
    v8f D01 = __builtin_amdgcn_wmma_f32_16x16x4_f32(false, A0, false, B1, (short)0, c1, false, false);
    v8f D10 = __builtin_amdgcn_wmma_f32_16x16x4_f32(false, A1, false, B0, (short)0, c0, false, false);
    v8f D11 = __builtin_amdgcn_wmma_f32_16x16x4_f32(false, A1, false, B1, (short)0, c1, false, false);

    // Branchless argmin on clamped d^2 (sqrt monotonic -> same argmin as
    // reference's dist argmin). Sortable u64 key: nonnegative floats order as
    // their bit patterns, so (bits(d2) << 32) | flat gives lexicographic
    // (value, smallest-flat-index) min -- exactly jnp.argmin's tie-break.
    unsigned long long best = ~0ull;
    const int hiOff = hi ? 8 : 0;

    auto consider = [&](float d2v, int a, int b) {
        float d = fmaxf(d2v, 0.0f);
        unsigned long long key =
            ((unsigned long long)f32_bits(d) << 32) | (unsigned int)((a << 5) | b);
        best = (key < best) ? key : best;
    };

    #pragma unroll
    for (int r = 0; r < 8; ++r) {
        const int a = r + hiOff;           // M within tile
        consider(D00[r], a,      n);
        consider(D01[r], a,      16 + n);
        consider(D10[r], 16 + a, n);
        consider(D11[r], 16 + a, 16 + n);
    }

    // cross-lane butterfly min of the packed keys
    #pragma unroll
    for (int m = 16; m >= 1; m >>= 1) {
        unsigned long long o = __shfl_xor(best, m, 32);
        best = (o < best) ? o : best;
    }

    // features for the winning voxel pair
    const int flat = (int)(best & 0xFFFFFFFFull);
    const int a = flat >> 5;
    const int b = flat & 31;
    const float4 p1 = dv[ci * S_PER_CLUST + a];
    const float4 p2 = dv[cj * S_PER_CLUST + b];
    const float lend  = __builtin_sqrtf(bits_f32((unsigned int)(best >> 32)));
    const float denom = (lend == 0.0f) ? 1.0f : lend;  // lend + (lend==0)
    const float dx = (p1.x - p2.x) / denom;
    const float dy = (p1.y - p2.y) / denom;
    const float dz = (p1.z - p2.z) / denom;

    float f = 0.0f;
    switch (lane) {
        case 0:  f = p1.x;  break;
        case 1:  f = p1.y;  break;
        case 2:  f = p1.z;  break;
        case 3:  f = p2.x;  break;
        case 4:  f = p2.y;  break;
        case 5:  f = p2.z;  break;
        case 6:  f = dx;    break;
        case 7:  f = dy;    break;
        case 8:  f = dz;    break;
        case 9:  f = lend;  break;
        case 10: f = dx*dx; break;
        case 11: f = dx*dy; break;
        case 12: f = dx*dz; break;
        case 13: f = dy*dx; break;
        case 14: f = dy*dy; break;
        case 15: f = dy*dz; break;
        case 16: f = dz*dx; break;
        case 17: f = dz*dy; break;
        case 18: f = dz*dz; break;
        default: break;
    }
    if (lane < 19) out[(size_t)e * 19 + lane] = f;
}

extern "C" void kernel_launch(void* const* d_in, const int* in_sizes, int n_in,
                              void* d_out, int out_size, void* d_ws, size_t ws_size,
                              hipStream_t stream)
{
    (void)n_in; (void)out_size; (void)d_ws; (void)ws_size;
    const float* data = (const float*)d_in[0];        // [N_VOX, 4] f32
    const int*   edge = (const int*)d_in[1];          // [2, E] int32
    float*       out  = (float*)d_out;                // [E, 19] f32

    const int n_edges = in_sizes[1] / 2;
    const int blocks  = (n_edges + 7) / 8;            // 8 waves (edges) per 256-thread block
    clust_geo_edge_wmma<<<blocks, 256, 0, stream>>>(data, edge, out, n_edges);
}